// QuestionAnswerModel_58471684768012
// MI455X (gfx1250) — compile-verified
//
#include <hip/hip_runtime.h>
#include <stdint.h>

#define MARGIN_F   6.0f
#define DIMD       256
#define TILE_E     16                 // entities per LDS tile (32 KB/tile)
#define TPB_TILES  8                  // tiles handled per block
#define THREADS    256                // 8 waves (wave32)
// phase in *revolutions*: phase_rad/2pi = r * (pi*DIM/(margin+eps)) / (2pi) = 16*r
#define PHASE_REV  16.0f

// Async-copy one 2048-byte entity row (512 floats) from global into LDS.
// Each lane moves 16B per instruction; 4 instructions cover 2 KB (32 lanes x 16B x 4).
__device__ __forceinline__ void async_row_2k(const float* __restrict__ gsrc,
                                             unsigned lds_off, int lane) {
  unsigned long long g = (unsigned long long)(uintptr_t)gsrc + (unsigned)(lane * 16);
  unsigned l = lds_off + (unsigned)(lane * 16);
#pragma unroll
  for (int j = 0; j < 4; ++j) {
    asm volatile("global_load_async_to_lds_b128 %0, %1, off"
                 :: "v"(l), "v"(g) : "memory");
    g += 512ull;
    l += 512u;
  }
}

__global__ __launch_bounds__(THREADS, 1)
void rotate_score_kernel(const float* __restrict__ head,
                         const float* __restrict__ relation,
                         const float* __restrict__ ent,
                         float* __restrict__ out,
                         int E, int tiles_total) {
  __shared__ float s_buf[2][TILE_E * 2 * DIMD];   // 2 x 32 KB double buffer
  __shared__ float s_partial[TILE_E * 256];       // 16 KB cross-wave partials

  const int tid  = threadIdx.x;
  const int lane = tid & 31;   // batch index (wave32: lane == batch)
  const int w    = tid >> 5;   // wave id 0..7, owns d-chunk [32w, 32w+32)

  // ---- per-lane rotated head for this wave's d-chunk, held in VGPRs ----
  // v_sin_f32/v_cos_f32 take revolutions (D = sin(2*pi*S0), hw fract): rev = 16*r exactly.
  float rre[32], rim[32];
  {
    const float* hrow = head + lane * (2 * DIMD);
    const float* rrow = relation + lane * DIMD;
#pragma unroll
    for (int i = 0; i < 32; ++i) {
      const int d = w * 32 + i;
      const float reh = hrow[d];
      const float imh = hrow[DIMD + d];
      const float rev = rrow[d] * PHASE_REV;
      const float sn = __builtin_amdgcn_sinf(rev);
      const float cs = __builtin_amdgcn_cosf(rev);
      rre[i] = reh * cs - imh * sn;
      rim[i] = reh * sn + imh * cs;
    }
  }

  // LDS byte offset of the buffers (flat->LDS aperture: low 32 bits are the offset)
  const unsigned buf_off0 = (unsigned)(uintptr_t)(&s_buf[0][0]);
  const int tile0 = blockIdx.x * TPB_TILES;

  auto issue_tile = [&](int t, int bufi) {
    // wave w copies entity rows 2w and 2w+1 of the 16-row tile
#pragma unroll
    for (int r = 0; r < 2; ++r) {
      int e = t * TILE_E + w * 2 + r;
      if (e >= E) e = E - 1;                       // clamp (dup rows, discarded on store)
      const float* src = ent + (size_t)e * (2 * DIMD);
      const unsigned loff = buf_off0
                          + (unsigned)bufi * (TILE_E * 2 * DIMD * 4)
                          + (unsigned)(w * 2 + r) * (2 * DIMD * 4);
      async_row_2k(src, loff, lane);
    }
  };

  // ---- prologue: stage tile 0 ----
  if (tile0 < tiles_total) issue_tile(tile0, 0);
  asm volatile("s_wait_asynccnt 0" ::: "memory");
  __syncthreads();

  for (int tt = 0; tt < TPB_TILES; ++tt) {
    const int t = tile0 + tt;
    if (t >= tiles_total) break;                   // block-uniform
    const int cur = tt & 1;
    const int nxt = cur ^ 1;
    if ((tt + 1 < TPB_TILES) && (t + 1 < tiles_total)) issue_tile(t + 1, nxt);

    // ---- compute: wave w handles its 32-dim chunk for all 16 entities ----
#pragma unroll 2
    for (int e = 0; e < TILE_E; ++e) {
      float a = 0.0f;
      const int rb = e * (2 * DIMD) + w * 32;
#pragma unroll
      for (int q = 0; q < 8; ++q) {
        const float4 tr = *(const float4*)(&s_buf[cur][rb + q * 4]);          // broadcast ds_load_b128
        const float4 ti = *(const float4*)(&s_buf[cur][rb + DIMD + q * 4]);
        float dr, di;
        dr = rre[q * 4 + 0] - tr.x; di = rim[q * 4 + 0] - ti.x;
        a += __builtin_amdgcn_sqrtf(__builtin_fmaf(di, di, dr * dr));
        dr = rre[q * 4 + 1] - tr.y; di = rim[q * 4 + 1] - ti.y;
        a += __builtin_amdgcn_sqrtf(__builtin_fmaf(di, di, dr * dr));
        dr = rre[q * 4 + 2] - tr.z; di = rim[q * 4 + 2] - ti.z;
        a += __builtin_amdgcn_sqrtf(__builtin_fmaf(di, di, dr * dr));
        dr = rre[q * 4 + 3] - tr.w; di = rim[q * 4 + 3] - ti.w;
        a += __builtin_amdgcn_sqrtf(__builtin_fmaf(di, di, dr * dr));
      }
      s_partial[e * 256 + w * 32 + lane] = a;      // bank-conflict-free
    }
    __syncthreads();

    // ---- cross-wave reduction + store ----
    {
      const int b  = tid >> 3;    // batch 0..31
      const int eo = tid & 7;
#pragma unroll
      for (int k = 0; k < 4; ++k) {
        const int e = eo + 8 * k;
        float ssum = 0.0f;
#pragma unroll
        for (int ww = 0; ww < 8; ++ww) ssum += s_partial[e * 256 + ww * 32 + b];
        const long long eg = (long long)t * TILE_E + e;
        if (eg < (long long)E)
          out[(size_t)b * (size_t)E + (size_t)eg] = MARGIN_F - ssum;
      }
    }

    asm volatile("s_wait_asynccnt 0" ::: "memory");  // next buffer resident
    __syncthreads();
  }
}

extern "C" void kernel_launch(void* const* d_in, const int* in_sizes, int n_in,
                              void* d_out, int out_size, void* d_ws, size_t ws_size,
                              hipStream_t stream) {
  (void)n_in; (void)out_size; (void)d_ws; (void)ws_size;
  const float* head     = (const float*)d_in[0];
  const float* relation = (const float*)d_in[1];
  const float* ent      = (const float*)d_in[2];
  float* out            = (float*)d_out;

  const int E     = in_sizes[2] / (2 * DIMD);            // 43234
  const int tiles = (E + TILE_E - 1) / TILE_E;           // 2703
  const int nb    = (tiles + TPB_TILES - 1) / TPB_TILES; // 338

  hipLaunchKernelGGL(rotate_score_kernel, dim3(nb), dim3(THREADS), 0, stream,
                     head, relation, ent, out, E, tiles);
}